// DKSCombiner_51960514347229
// MI455X (gfx1250) — compile-verified
//
#include <hip/hip_runtime.h>
#include <math.h>

typedef float v2f __attribute__((ext_vector_type(2)));
typedef float v8f __attribute__((ext_vector_type(8)));

#define B_  128
#define T_  2048
#define H_  256
#define L_  64
#define MB  16      // batch rows per workgroup == WMMA M

static __device__ __forceinline__ v8f splat8(float x) {
  v8f v;
#pragma unroll
  for (int i = 0; i < 8; ++i) v[i] = x;
  return v;
}

// Branch-free fast transcendentals: lower to v_exp_f32 / v_log_f32 /
// v_rcp_f32 / v_sqrt_f32 TRANS ops (co-execute with the WMMA pipe, no
// EXEC-masked slow paths on the scan's serial chain).
static __device__ __forceinline__ float tanh_fast(float x) {
  // tanh(x) = 1 - 2 / (exp(2x) + 1)
  float e = __builtin_amdgcn_exp2f(x * 2.885390081777926f);   // exp(2x)
  return 1.0f - 2.0f * __builtin_amdgcn_rcpf(e + 1.0f);
}

static __device__ __forceinline__ float softplus_fast(float x) {
  // softplus(x) = max(x,0) + ln2 * log2(1 + exp(-|x|))   (stable both signs)
  float e = __builtin_amdgcn_exp2f(-fabsf(x) * 1.4426950408889634f);
  float l = __builtin_amdgcn_logf(1.0f + e) * 0.6931471805599453f;
  return fmaxf(x, 0.0f) + l;
}

static __device__ __forceinline__ float sqrt_fast(float x) {
  return __builtin_amdgcn_sqrtf(x);
}

__global__ __launch_bounds__(256, 1)
void dks_combiner_scan(const float* __restrict__ hr,    // [B,T,H]
                       const float* __restrict__ eps,   // [B,T,L]
                       const float* __restrict__ Wc,    // [H,L]
                       const float* __restrict__ bc,    // [H]
                       const float* __restrict__ Wmu,   // [L,H]
                       const float* __restrict__ bmu,   // [L]
                       const float* __restrict__ Ws,    // [L,H]
                       const float* __restrict__ bs,    // [L]
                       float* __restrict__ out)         // Z | mu | var, each [B,T,L]
{
  // Weights live in LDS for the entire 2048-step scan (needs CDNA5's 320KB WGP LDS).
  __shared__ float sWcT [L_ * H_];   // [k=64 ][n=256]  = Wc^T
  __shared__ float sWmuT[H_ * L_];   // [k=256][n=64 ]  = Wmu^T
  __shared__ float sWsT [H_ * L_];   // [k=256][n=64 ]  = Ws^T
  __shared__ float s_hc [MB * H_];   // [16][256]
  __shared__ float s_z  [MB * L_];   // [16][64]
  __shared__ float s_var[MB * L_];   // [16][64]
  __shared__ float s_bc [H_];
  __shared__ float s_bmu[L_];
  __shared__ float s_bs [L_];

  const int tid  = threadIdx.x;
  const int wave = tid >> 5;
  const int lane = tid & 31;
  const int lo   = lane & 15;   // N (or M for A-operand) index inside tile
  const int hi   = lane >> 4;   // selects K pair / row half per ISA layout
  const int b0   = blockIdx.x * MB;

  // ---------------- one-time LDS init ----------------
  for (int i = tid; i < H_ * L_; i += 256) {      // Wc[n][k] -> WcT[k][n]
    int n = i / L_, k = i % L_;
    sWcT[k * H_ + n] = Wc[i];
  }
  for (int i = tid; i < L_ * H_; i += 256) {      // Wmu[l][h] -> WmuT[h][l]
    int l = i / H_, h = i % H_;
    sWmuT[h * L_ + l] = Wmu[i];
    sWsT [h * L_ + l] = Ws[i];
  }
  if (tid < H_) s_bc[tid] = bc[tid];
  if (tid < L_) { s_bmu[tid] = bmu[tid]; s_bs[tid] = bs[tid]; }
  for (int i = tid; i < MB * L_; i += 256) s_z[i] = 0.0f;   // z0 = 0
  __syncthreads();

  const size_t BTL = (size_t)B_ * T_ * L_;
  float* __restrict__ outZ   = out;
  float* __restrict__ outMu  = out + BTL;
  float* __restrict__ outVar = out + 2 * BTL;

  const bool  isMu = (wave < 4);
  const int   n0s  = (isMu ? wave : (wave - 4)) * 16;   // small-GEMM N tile base
  const float* Wt  = isMu ? sWmuT : sWsT;
  const float bias2 = isMu ? s_bmu[n0s + lo] : s_bs[n0s + lo];

  for (int t = 0; t < T_; ++t) {
    // ---------- GEMM1: hc_pre[16,256] = z[16,64] @ WcT + bc ; 2 N-tiles per wave ----------
#pragma unroll
    for (int tt = 0; tt < 2; ++tt) {
      const int nt = wave * 2 + tt;
      const int n  = nt * 16 + lo;
      v8f acc0 = splat8(s_bc[n]);      // bias folded into accumulator
      v8f acc1 = splat8(0.0f);
#pragma unroll
      for (int kt = 0; kt < 16; ++kt) {
        const int kb = kt * 4 + 2 * hi;                    // k for VGPR0; VGPR1 is k+1
        v2f a = { s_z[lo * L_ + kb], s_z[lo * L_ + kb + 1] };
        v2f b = { sWcT[(kb    ) * H_ + n],
                  sWcT[(kb + 1) * H_ + n] };
        if (kt & 1)
          acc1 = __builtin_amdgcn_wmma_f32_16x16x4_f32(false, a, false, b,
                                                       (short)0, acc1, false, false);
        else
          acc0 = __builtin_amdgcn_wmma_f32_16x16x4_f32(false, a, false, b,
                                                       (short)0, acc0, false, false);
      }
      v8f acc = acc0 + acc1;
      // activation + average with streamed h_right, stash hc into LDS
#pragma unroll
      for (int r = 0; r < 8; ++r) {
        const int m  = r + 8 * hi;
        const int bg = b0 + m;
        const float hrv = hr[((size_t)bg * T_ + t) * H_ + n];
        s_hc[m * H_ + n] = 0.5f * (tanh_fast(acc[r]) + hrv);
      }
    }
    __syncthreads();   // hc ready

    // ---------- GEMM2/3: [16,256] @ [256,64]; waves 0-3 -> mu, waves 4-7 -> var ----------
    {
      v8f acc[4];
      acc[0] = splat8(bias2);
      acc[1] = splat8(0.0f); acc[2] = splat8(0.0f); acc[3] = splat8(0.0f);
#pragma unroll
      for (int kt = 0; kt < 64; ++kt) {
        const int kb = kt * 4 + 2 * hi;
        v2f a = { s_hc[lo * H_ + kb], s_hc[lo * H_ + kb + 1] };
        v2f b = { Wt[(kb    ) * L_ + n0s + lo],
                  Wt[(kb + 1) * L_ + n0s + lo] };
        acc[kt & 3] = __builtin_amdgcn_wmma_f32_16x16x4_f32(false, a, false, b,
                                                            (short)0, acc[kt & 3],
                                                            false, false);
      }
      v8f res = (acc[0] + acc[1]) + (acc[2] + acc[3]);

      if (!isMu) {
        // variance = softplus(pre); publish to LDS + global
#pragma unroll
        for (int r = 0; r < 8; ++r) {
          const int m  = r + 8 * hi;
          const int bg = b0 + m;
          const float vv = softplus_fast(res[r]);
          const size_t oi = ((size_t)bg * T_ + t) * L_ + n0s + lo;
          s_var[m * L_ + n0s + lo] = vv;
          outVar[oi] = vv;
        }
      }
      __syncthreads();  // var ready

      if (isMu) {
#pragma unroll
        for (int r = 0; r < 8; ++r) {
          const int m  = r + 8 * hi;
          const int bg = b0 + m;
          const size_t oi = ((size_t)bg * T_ + t) * L_ + n0s + lo;
          const float mu = res[r];
          const float vv = s_var[m * L_ + n0s + lo];
          const float zv = mu + sqrt_fast(vv) * eps[oi];
          s_z[m * L_ + n0s + lo] = zv;   // feedback for next step
          outZ[oi]  = zv;
          outMu[oi] = mu;
        }
      } else if (t + 1 < T_) {
        // waves 4-7: prefetch next timestep's h_right tile (16 rows x 1KB = 128 lines)
        const int li   = (wave - 4) * 32 + lane;      // 0..127
        const int row  = li >> 3;                     // 0..15
        const int seg  = li & 7;                      // 128B line within row
        const int bg   = b0 + row;
        const float* p = &hr[((size_t)bg * T_ + (t + 1)) * H_ + seg * 32];
        __builtin_prefetch((const void*)p, 0, 1);
      }
    }
    __syncthreads();   // z ready for next iteration
  }
}

extern "C" void kernel_launch(void* const* d_in, const int* in_sizes, int n_in,
                              void* d_out, int out_size, void* d_ws, size_t ws_size,
                              hipStream_t stream) {
  (void)in_sizes; (void)n_in; (void)out_size; (void)d_ws; (void)ws_size;
  const float* hr  = (const float*)d_in[0];
  const float* eps = (const float*)d_in[1];
  const float* Wc  = (const float*)d_in[2];
  const float* bc  = (const float*)d_in[3];
  const float* Wmu = (const float*)d_in[4];
  const float* bmu = (const float*)d_in[5];
  const float* Ws  = (const float*)d_in[6];
  const float* bs  = (const float*)d_in[7];
  float* out = (float*)d_out;

  dks_combiner_scan<<<dim3(B_ / MB), dim3(256), 0, stream>>>(
      hr, eps, Wc, bc, Wmu, bmu, Ws, bs, out);
}